// RegionLoss_3D_kl_61263413510184
// MI455X (gfx1250) — compile-verified
//
#include <hip/hip_runtime.h>

// RegionLoss 3D: avg-pool(12^3 blocks) -> gram/cosine-sim over 512 tokens x 32 ch
// -> MSE between the two similarity matrices. fp32 throughout (matches reference).
//
// MI455X reasoning: 453 MB of input @ 23.3 TB/s ~= 19.4 us floor; pooling is the
// whole cost, so it uses aligned float4 streaming loads. Pooled features (256 KB)
// are L2-resident. Gram matrices use V_WMMA_F32_16X16X4_F32 (fp32 WMMA); all
// fragment loads are hoisted ahead of the WMMA chain so the loads issue as one
// clause and the matrix ops drain with staged s_wait_loadcnt values instead of
// a full stall per WMMA.

typedef __attribute__((ext_vector_type(2))) float v2f;
typedef __attribute__((ext_vector_type(8))) float v8f;

#define SS    8            // pooled grid per axis
#define LTOK  512          // SS^3 tokens
#define CCH   32           // channels (== wave32 lanes)
#define BATCH 2
#define DIM   96
#define BLK   12           // DIM / SS
#define EPSV  1e-8f

__device__ __forceinline__ float wave_reduce_sum(float v) {
    v += __shfl_xor(v, 1, 32);
    v += __shfl_xor(v, 2, 32);
    v += __shfl_xor(v, 4, 32);
    v += __shfl_xor(v, 8, 32);
    v += __shfl_xor(v, 16, 32);
    return v;
}

// ---------------------------------------------------------------------------
// 1) Pooling: one wave per pooled cell. 1728 elems = 144 rows x 12 floats.
//    Each 12-float run is 48 B and 16 B-aligned -> 3 float4 loads per row,
//    432 float4 per cell striped across 32 lanes (fully coalesced).
// ---------------------------------------------------------------------------
__global__ void pool_kernel(const float* __restrict__ p1,
                            const float* __restrict__ p2,
                            float* __restrict__ fa,
                            float* __restrict__ fb) {
    const int wave = (blockIdx.x * blockDim.x + threadIdx.x) >> 5;   // [0, 65536)
    const int lane = threadIdx.x & 31;

    const int tensor = wave >> 15;          // 0 -> p1, 1 -> p2
    const int cell   = wave & 32767;        // b*C*L + c*L + l
    const int b   = cell >> 14;             // / (32*512)
    const int rem = cell & 16383;
    const int c   = rem >> 9;               // / 512
    const int l   = rem & 511;
    const int sd  = l >> 6;
    const int sh  = (l >> 3) & 7;
    const int sw  = l & 7;

    const float* __restrict__ src = tensor ? p2 : p1;
    const long base = (long)(b * CCH + c) * (DIM * DIM * DIM)
                    + (long)(sd * BLK) * (DIM * DIM)
                    + (long)(sh * BLK) * DIM
                    + (long)(sw * BLK);

    float sum = 0.f;
    for (int i = lane; i < 432; i += 32) {        // 432 float4 per cell
        const int row  = i / 3;                   // 0..143
        const int quad = i - row * 3;             // 0..2
        const int dd   = row / 12;
        const int hh   = row - dd * 12;
        const float4 v = *reinterpret_cast<const float4*>(
            src + base + (long)dd * (DIM * DIM) + (long)hh * DIM + quad * 4);
        sum += (v.x + v.y) + (v.z + v.w);
    }
    sum = wave_reduce_sum(sum);
    if (lane == 0) {
        float* __restrict__ dst = tensor ? fb : fa;
        dst[(b * LTOK + l) * CCH + c] = sum * (1.0f / (BLK * BLK * BLK));
    }
}

// ---------------------------------------------------------------------------
// 2) Token norms: C == 32 == wave32 lanes. One wave per (tensor, b, token).
// ---------------------------------------------------------------------------
__global__ void norm_kernel(const float* __restrict__ fa,
                            const float* __restrict__ fb,
                            float* __restrict__ na,
                            float* __restrict__ nb) {
    const int wave = (blockIdx.x * blockDim.x + threadIdx.x) >> 5;   // [0, 2048)
    const int lane = threadIdx.x & 31;                               // channel
    const int tensor = wave >> 10;
    const int bl     = wave & 1023;                                  // b*L + l
    const float* __restrict__ f = tensor ? fb : fa;
    const float v = f[bl * CCH + lane];
    const float s = wave_reduce_sum(v * v);
    if (lane == 0) (tensor ? nb : na)[bl] = sqrtf(s);
}

// ---------------------------------------------------------------------------
// 3) Gram tiles via V_WMMA_F32_16X16X4_F32.
//    One wave per (b, ti, tj) tile -> 2048 waves, 16 WMMAs each.
//    A layout (16x4 f32): lane l, vgpr v -> M = l%16, K = 2*(l/16) + v.
//    B layout (4x16 f32): lane l, vgpr v -> N = l%16, K = 2*(l/16) + v.
//    D layout (16x16 f32): lane l, vgpr v -> N = l%16, M = 8*(l/16) + v.
//    All 32 fragment loads are issued before the 16-WMMA chain.
// ---------------------------------------------------------------------------
__global__ void gram_mse_kernel(const float* __restrict__ fa,
                                const float* __restrict__ fb,
                                const float* __restrict__ na,
                                const float* __restrict__ nb,
                                float* __restrict__ partials) {
    const int wave = (blockIdx.x * blockDim.x + threadIdx.x) >> 5;   // [0, 2048)
    const int lane = threadIdx.x & 31;

    const int b  = wave >> 10;
    const int ti = (wave >> 5) & 31;
    const int tj = wave & 31;

    const int half = lane >> 4;      // 0 or 1
    const int lm   = lane & 15;
    const int i    = ti * 16 + lm;   // A-row this lane supplies
    const int j    = tj * 16 + lm;   // B-col this lane supplies

    const float* __restrict__ rowAi = fa + ((long)b * LTOK + i) * CCH + half * 2;
    const float* __restrict__ rowAj = fa + ((long)b * LTOK + j) * CCH + half * 2;
    const float* __restrict__ rowBi = fb + ((long)b * LTOK + i) * CCH + half * 2;
    const float* __restrict__ rowBj = fb + ((long)b * LTOK + j) * CCH + half * 2;

    // Hoist every fragment load: 4 streams x 8 chunks of 8B-aligned float2.
    v2f A1[8], B1[8], A2[8], B2[8];
#pragma unroll
    for (int kc = 0; kc < 8; ++kc) {
        A1[kc] = *reinterpret_cast<const v2f*>(rowAi + kc * 4);
        B1[kc] = *reinterpret_cast<const v2f*>(rowAj + kc * 4);
        A2[kc] = *reinterpret_cast<const v2f*>(rowBi + kc * 4);
        B2[kc] = *reinterpret_cast<const v2f*>(rowBj + kc * 4);
    }

    v8f acc_a = {};
    v8f acc_b = {};
#pragma unroll
    for (int kc = 0; kc < 8; ++kc) {
        acc_a = __builtin_amdgcn_wmma_f32_16x16x4_f32(
            false, A1[kc], false, B1[kc], (short)0, acc_a, false, false);
        acc_b = __builtin_amdgcn_wmma_f32_16x16x4_f32(
            false, A2[kc], false, B2[kc], (short)0, acc_b, false, false);
    }

    const float nja = na[b * LTOK + tj * 16 + lm];
    const float njb = nb[b * LTOK + tj * 16 + lm];

    float sum = 0.f;
#pragma unroll
    for (int v = 0; v < 8; ++v) {
        const int m = half * 8 + v;
        const float nia = na[b * LTOK + ti * 16 + m];
        const float nib = nb[b * LTOK + ti * 16 + m];
        const float sa = acc_a[v] / fmaxf(nia * nja, EPSV);
        const float sb = acc_b[v] / fmaxf(nib * njb, EPSV);
        const float d  = sa - sb;
        sum += d * d;
    }
    sum = wave_reduce_sum(sum);
    if (lane == 0) partials[wave] = sum;     // fixed slot -> deterministic
}

// ---------------------------------------------------------------------------
// 4) Deterministic final reduction: single wave, fixed order.
//    result = total / (B * L * L)  (per-sample mean over L^2, then batch mean)
// ---------------------------------------------------------------------------
__global__ void final_reduce_kernel(const float* __restrict__ partials,
                                    float* __restrict__ out) {
    const int lane = threadIdx.x;    // blockDim.x == 32
    float s = 0.f;
    for (int idx = lane; idx < 2048; idx += 32) s += partials[idx];
    s = wave_reduce_sum(s);
    if (lane == 0)
        out[0] = s * (1.0f / ((float)BATCH * (float)LTOK * (float)LTOK));
}

extern "C" void kernel_launch(void* const* d_in, const int* in_sizes, int n_in,
                              void* d_out, int out_size, void* d_ws, size_t ws_size,
                              hipStream_t stream) {
    const float* p1 = (const float*)d_in[0];
    const float* p2 = (const float*)d_in[1];

    float* ws = (float*)d_ws;
    float* fa = ws;                          // [2][512][32]  = 32768 floats
    float* fb = fa + BATCH * LTOK * CCH;     // [2][512][32]  = 32768 floats
    float* na = fb + BATCH * LTOK * CCH;     // [2][512]      =  1024 floats
    float* nb = na + BATCH * LTOK;           // [2][512]      =  1024 floats
    float* partials = nb + BATCH * LTOK;     // [2048]
    // total ws use: 69632 floats = 272 KB

    // 65536 pooled cells, 8 waves per 256-thread block
    pool_kernel<<<8192, 256, 0, stream>>>(p1, p2, fa, fb);
    // 2048 token norms
    norm_kernel<<<256, 256, 0, stream>>>(fa, fb, na, nb);
    // 2048 gram tiles (both tensors per wave), 16 wmma each
    gram_mse_kernel<<<256, 256, 0, stream>>>(fa, fb, na, nb, partials);
    // deterministic scalar reduction
    final_reduce_kernel<<<1, 32, 0, stream>>>(partials, (float*)d_out);
}